// AGCN_66425964200121
// MI455X (gfx1250) — compile-verified
//
#include <hip/hip_runtime.h>
#include <math.h>

// ---------------------------------------------------------------------------
// AGCN forward for MI455X (gfx1250), wave32, f32 WMMA (V_WMMA_F32_16X16X4_F32)
// B=32, N=1000 (padded 1024), Din=32, Dout=64, D=16, K=2, alpha=3
// Round 3: branch-free transcendentals. libm tanhf/expf expand to exec-mask
// branchy multi-path code; gfx1250 has native V_TANH_F32 / V_EXP_F32, so use
// __builtin_amdgcn_tanhf (guarded) and __expf to keep epilogues straight-line.
// ---------------------------------------------------------------------------

typedef float v2f __attribute__((ext_vector_type(2)));
typedef float v8f __attribute__((ext_vector_type(8)));

#define NP   1024
#define NN   1000
#define BB   32
#define DIN  32
#define DOUT 64
#define DD   16

// D = A(16x4, f32) * B(4x16, f32) + C(16x16, f32)
static __device__ __forceinline__ v8f wmma4(v2f a, v2f b, v8f c) {
  return __builtin_amdgcn_wmma_f32_16x16x4_f32(false, a, false, b, (short)0, c,
                                               false, false);
}

static __device__ __forceinline__ v2f ld2(const float* p) {
  return *(const v2f*)p;
}

// Branch-free tanh: native V_TANH_F32 on gfx1250, else exp-based identity
// (single v_exp_f32, no control flow).
static __device__ __forceinline__ float fast_tanh(float x) {
#if __has_builtin(__builtin_amdgcn_tanhf)
  return __builtin_amdgcn_tanhf(x);
#else
  float e = __expf(-2.0f * x);
  return (1.0f - e) / (1.0f + e);
#endif
}

static __device__ __forceinline__ float relutanh3(float v) {
  float t = fast_tanh(3.0f * v);
  return t > 0.0f ? t : 0.0f;
}

static __device__ __forceinline__ float fast_sigmoid(float z) {
  return 1.0f / (1.0f + __expf(-z));
}

// ---------------------------------------------------------------------------
// P1: Epad[NP][DD] = E zero-padded past row NN.
// ---------------------------------------------------------------------------
__global__ void k_pade(const float* __restrict__ E, float* __restrict__ Epad) {
  int i = blockIdx.x * 256 + threadIdx.x;       // NP*DD = 16384
  int r = i >> 4;
  int c = i & 15;
  Epad[i] = (r < NN) ? E[r * DD + c] : 0.0f;
}

// ---------------------------------------------------------------------------
// P2: xpad[BB][NP][DIN] = x zero-padded past node NN.
// ---------------------------------------------------------------------------
__global__ void k_padx(const float* __restrict__ x, float* __restrict__ xpad) {
  int i = blockIdx.x * 256 + threadIdx.x;       // BB*NP*DIN = 1048576
  int c = i & (DIN - 1);
  int r = (i >> 5) & (NP - 1);
  int b = i >> 15;                              // NP*DIN = 2^15
  xpad[i] = (r < NN) ? x[(b * NN + r) * DIN + c] : 0.0f;
}

// ---------------------------------------------------------------------------
// K1: AS = relu(tanh(3 * E E^T)), padded NP x NP (pads exactly zero since
// Epad pad rows are zero -> dot 0 -> relu(tanh 0) = 0). One wave per tile.
// ---------------------------------------------------------------------------
__global__ void k_as(const float* __restrict__ Epad, float* __restrict__ AS) {
  int lane = threadIdx.x & 31;
  int lid = lane & 15, half = lane >> 4;
  int tm = blockIdx.x, tn = blockIdx.y;
  int ra = tm * 16 + lid;
  int rb = tn * 16 + lid;
  v8f acc = {};
#pragma unroll
  for (int kk = 0; kk < 4; ++kk) {
    int k0 = kk * 4 + 2 * half;
    v2f a = ld2(&Epad[ra * DD + k0]);
    v2f b = ld2(&Epad[rb * DD + k0]);
    acc = wmma4(a, b, acc);
  }
#pragma unroll
  for (int v = 0; v < 8; ++v) {
    int row = tm * 16 + v + 8 * half;
    int col = tn * 16 + lid;
    AS[row * NP + col] = relutanh3(acc[v]);
  }
}

// ---------------------------------------------------------------------------
// K2: d[n] = rowsum(AS[n,:])^-0.5 (0 for pad rows).
// ---------------------------------------------------------------------------
__global__ void k_rowsum(const float* __restrict__ AS, float* __restrict__ dvec) {
  __shared__ float red[256];
  int row = blockIdx.x;
  int t = threadIdx.x;
  float s = 0.0f;
  for (int c = t; c < NP; c += 256) s += AS[row * NP + c];
  red[t] = s;
  __syncthreads();
  for (int off = 128; off > 0; off >>= 1) {
    if (t < off) red[t] += red[t + off];
    __syncthreads();
  }
  if (t == 0) dvec[row] = (row < NN) ? rsqrtf(red[0]) : 0.0f;
}

// ---------------------------------------------------------------------------
// K3: one K=NP pass, two accumulators (AS is symmetric):
//   ho[b,n,c]     = sum_v AS[n,v] * x[b,v,c]
//   xag[b,n,32+c] = d[n] * sum_v AS[n,v] * d[v] * x[b,v,c]   (= AG @ x)
//   xag[b,n,c]    = x[b,n,c]
// ---------------------------------------------------------------------------
__global__ void k_asx(const float* __restrict__ AS, const float* __restrict__ xpad,
                      const float* __restrict__ dvec,
                      float* __restrict__ ho, float* __restrict__ xag) {
  int lane = threadIdx.x & 31;
  int lid = lane & 15, half = lane >> 4;
  int tm = blockIdx.x;            // row tile (NP/16)
  int tn = blockIdx.y;            // col tile over DIN (2)
  int b  = blockIdx.z;            // batch
  int arow = tm * 16 + lid;
  int col  = tn * 16 + lid;
  const float* asrow = AS + (size_t)arow * NP;
  const float* xb = xpad + (size_t)b * NP * DIN;
  v8f ah = {}, ar = {};
#pragma unroll 8
  for (int kt = 0; kt < NP / 4; ++kt) {
    int k0 = kt * 4 + 2 * half;
    v2f a  = ld2(&asrow[k0]);
    v2f dv = ld2(&dvec[k0]);
    v2f bx, bdx;
    bx[0] = xb[k0 * DIN + col];
    bx[1] = xb[(k0 + 1) * DIN + col];
    bdx[0] = dv[0] * bx[0];
    bdx[1] = dv[1] * bx[1];
    ah = wmma4(a, bx, ah);
    ar = wmma4(a, bdx, ar);
  }
#pragma unroll
  for (int v = 0; v < 8; ++v) {
    int row = tm * 16 + v + 8 * half;
    float dn = dvec[row];         // 0 on pad rows
    ho[(b * NP + row) * DIN + col] = ah[v];
    xag[(b * NP + row) * (2 * DIN) + DIN + col] = dn * ar[v];
    xag[(b * NP + row) * (2 * DIN) + col] = xb[row * DIN + col];
  }
}

// ---------------------------------------------------------------------------
// K4: per-(b,n) wave: sigmoid MLP, DF, Et, DE = tanh(3*Et*DF).
// Pad rows write DE = 0 so the fused DA pass naturally ignores them.
// ---------------------------------------------------------------------------
__global__ void k_mlp(const float* __restrict__ ho, const float* __restrict__ E,
                      const float* __restrict__ TD, const float* __restrict__ TW,
                      const float* __restrict__ fc1w, const float* __restrict__ fc1b,
                      const float* __restrict__ fc2w, const float* __restrict__ fc2b,
                      const float* __restrict__ fc3w, const float* __restrict__ fc3b,
                      float* __restrict__ DE) {
  __shared__ float sho[DIN];
  __shared__ float sh[DIN];
  __shared__ float sh2[DD];
  int n = blockIdx.x, b = blockIdx.y;
  int t = threadIdx.x;
  if (n >= NN) {
    if (t < DD) DE[(b * NP + n) * DD + t] = 0.0f;
    return;
  }
  sho[t] = ho[(b * NP + n) * DIN + t];
  __syncthreads();
  float z = fc1b[t];
#pragma unroll
  for (int i = 0; i < DIN; ++i) z += sho[i] * fc1w[t * DIN + i];
  sh[t] = fast_sigmoid(z);
  __syncthreads();
  if (t < DD) {
    float z2 = fc2b[t];
#pragma unroll
    for (int i = 0; i < DIN; ++i) z2 += sh[i] * fc2w[t * DIN + i];
    sh2[t] = fast_sigmoid(z2);
  }
  __syncthreads();
  if (t < DD) {
    float df = fc3b[t];
#pragma unroll
    for (int i = 0; i < DD; ++i) df += sh2[i] * fc3w[t * DD + i];
    float et = E[n * DD + t] * TD[(b * NN + n) * DD + t] * TW[(b * NN + n) * DD + t];
    DE[(b * NP + n) * DD + t] = fast_tanh(3.0f * et * df);
  }
}

// ---------------------------------------------------------------------------
// K5: fused DA @ x (flash-style): per (b, query tile) wave, loop over m-tiles:
//   S = DE_q @ DE_m^T (WMMA), S = relu(tanh(3S)), LDS bounce D->A layout,
//   acc += S @ x_m (WMMA). Never materializes the 128MB DA.
// ---------------------------------------------------------------------------
__global__ void k_dax(const float* __restrict__ DE, const float* __restrict__ xpad,
                      float* __restrict__ xdg2) {
  __shared__ float sS[16 * 16];
  int lane = threadIdx.x & 31;
  int lid = lane & 15, half = lane >> 4;
  int tq = blockIdx.x, b = blockIdx.y;
  const float* deb = DE + (size_t)b * NP * DD;
  const float* xb  = xpad + (size_t)b * NP * DIN;
  v2f aDE[4];
#pragma unroll
  for (int kk = 0; kk < 4; ++kk) {
    int k0 = kk * 4 + 2 * half;
    aDE[kk] = ld2(&deb[(tq * 16 + lid) * DD + k0]);
  }
  v8f acc0 = {}, acc1 = {};
  for (int mt = 0; mt < NP / 16; ++mt) {
    v8f s = {};
#pragma unroll
    for (int kk = 0; kk < 4; ++kk) {
      int k0 = kk * 4 + 2 * half;
      v2f bd = ld2(&deb[(mt * 16 + lid) * DD + k0]);
      s = wmma4(aDE[kk], bd, s);
    }
    __syncthreads();                       // WAR on sS vs previous iteration
#pragma unroll
    for (int v = 0; v < 8; ++v)
      sS[(v + 8 * half) * 16 + lid] = relutanh3(s[v]);
    __syncthreads();
#pragma unroll
    for (int kk = 0; kk < 4; ++kk) {
      int k0 = kk * 4 + 2 * half;
      v2f a2 = *(const v2f*)&sS[lid * 16 + k0];
      int m0 = mt * 16 + k0;
      v2f b0, b1;
      b0[0] = xb[m0 * DIN + lid];
      b0[1] = xb[(m0 + 1) * DIN + lid];
      b1[0] = xb[m0 * DIN + 16 + lid];
      b1[1] = xb[(m0 + 1) * DIN + 16 + lid];
      acc0 = wmma4(a2, b0, acc0);
      acc1 = wmma4(a2, b1, acc1);
    }
  }
#pragma unroll
  for (int v = 0; v < 8; ++v) {
    int row = tq * 16 + v + 8 * half;
    xdg2[(b * NP + row) * DIN + lid]      = acc0[v];
    xdg2[(b * NP + row) * DIN + 16 + lid] = acc1[v];
  }
}

// ---------------------------------------------------------------------------
// K6: per node n (grid = 1000): build W_n[64x64] = sum_d E[n,d]*Wp[d] and
// bias[n,:] in LDS, then 8 waves compute
//   out[:,n,:] = a * (Xag(32x64) @ W_n + bias) + b * (Xdg2(32x32) @ Wdg)
// ---------------------------------------------------------------------------
__global__ void k_final(const float* __restrict__ xag, const float* __restrict__ xdg2,
                        const float* __restrict__ E, const float* __restrict__ Wp,
                        const float* __restrict__ biasp, const float* __restrict__ Wdg,
                        const float* __restrict__ pa, const float* __restrict__ pb,
                        float* __restrict__ out) {
  __shared__ float sW[64 * 64];
  __shared__ float sbias[64];
  __shared__ float sE[DD];
  int n = blockIdx.x;
  int t = threadIdx.x;
  if (t < DD) sE[t] = E[n * DD + t];
  __syncthreads();
  for (int e = t; e < 64 * 64; e += 256) {
    int ki = e >> 6, o = e & 63;
    float s = 0.0f;
#pragma unroll
    for (int d0 = 0; d0 < DD; ++d0) s += sE[d0] * Wp[(d0 * 64 + ki) * 64 + o];
    sW[e] = s;                              // sW[ki][o]
  }
  if (t < 64) {
    float s = 0.0f;
#pragma unroll
    for (int d0 = 0; d0 < DD; ++d0) s += sE[d0] * biasp[d0 * 64 + t];
    sbias[t] = s;
  }
  __syncthreads();

  int lane = t & 31, w = t >> 5;
  int lid = lane & 15, half = lane >> 4;
  int mt = w & 1, nt = w >> 1;              // 2 row tiles (b) x 4 col tiles (o)
  int brow = mt * 16 + lid;                 // batch index for A loads
  int oc = nt * 16 + lid;                   // output channel for B loads
  const float* aga = xag + ((size_t)brow * NP + n) * (2 * DIN);
  const float* dga = xdg2 + ((size_t)brow * NP + n) * DIN;
  v8f accA = {}, accD = {};
#pragma unroll
  for (int kk = 0; kk < 16; ++kk) {         // ag path: K = 64
    int k0 = kk * 4 + 2 * half;
    v2f a = ld2(&aga[k0]);
    v2f bm;
    bm[0] = sW[k0 * 64 + oc];
    bm[1] = sW[(k0 + 1) * 64 + oc];
    accA = wmma4(a, bm, accA);
  }
#pragma unroll
  for (int kk = 0; kk < 8; ++kk) {          // dg path: K = 32
    int k0 = kk * 4 + 2 * half;
    v2f a = ld2(&dga[k0]);
    v2f bm;
    bm[0] = Wdg[k0 * DOUT + oc];
    bm[1] = Wdg[(k0 + 1) * DOUT + oc];
    accD = wmma4(a, bm, accD);
  }
  float av = pa[0], bv = pb[0];
  float bo = sbias[oc];
#pragma unroll
  for (int v = 0; v < 8; ++v) {
    int bb = mt * 16 + v + 8 * half;        // batch row of this D element
    out[(bb * NN + n) * DOUT + oc] = av * (accA[v] + bo) + bv * accD[v];
  }
}

// ---------------------------------------------------------------------------
// Launch. Workspace (floats):
//   AS   : NP*NP       = 1,048,576
//   dvec : NP          =     1,024
//   ho   : BB*NP*DIN   = 1,048,576
//   xag  : BB*NP*2*DIN = 2,097,152
//   DE   : BB*NP*DD    =   524,288
//   xdg2 : BB*NP*DIN   = 1,048,576
//   Epad : NP*DD       =    16,384
//   xpad : BB*NP*DIN   = 1,048,576
// total ~6.83M floats = ~27.3 MB (fits in 192 MB L2).
// ---------------------------------------------------------------------------
extern "C" void kernel_launch(void* const* d_in, const int* in_sizes, int n_in,
                              void* d_out, int out_size, void* d_ws, size_t ws_size,
                              hipStream_t stream) {
  const float* x     = (const float*)d_in[0];
  const float* E     = (const float*)d_in[1];
  const float* TD    = (const float*)d_in[2];
  const float* TW    = (const float*)d_in[3];
  const float* Wp    = (const float*)d_in[4];
  const float* biasp = (const float*)d_in[5];
  const float* Wdg   = (const float*)d_in[6];
  const float* pa    = (const float*)d_in[7];
  const float* pb    = (const float*)d_in[8];
  const float* fc1w  = (const float*)d_in[9];
  const float* fc1b  = (const float*)d_in[10];
  const float* fc2w  = (const float*)d_in[11];
  const float* fc2b  = (const float*)d_in[12];
  const float* fc3w  = (const float*)d_in[13];
  const float* fc3b  = (const float*)d_in[14];

  float* ws   = (float*)d_ws;
  float* AS   = ws;
  float* dvec = AS + (size_t)NP * NP;
  float* ho   = dvec + NP;
  float* xag  = ho + (size_t)BB * NP * DIN;
  float* DE   = xag + (size_t)BB * NP * 2 * DIN;
  float* xdg2 = DE + (size_t)BB * NP * DD;
  float* Epad = xdg2 + (size_t)BB * NP * DIN;
  float* xpad = Epad + (size_t)NP * DD;

  k_pade  <<<(NP * DD) / 256, 256, 0, stream>>>(E, Epad);
  k_padx  <<<(BB * NP * DIN) / 256, 256, 0, stream>>>(x, xpad);
  k_as    <<<dim3(NP / 16, NP / 16), 32, 0, stream>>>(Epad, AS);
  k_rowsum<<<NP, 256, 0, stream>>>(AS, dvec);
  k_asx   <<<dim3(NP / 16, 2, BB), 32, 0, stream>>>(AS, xpad, dvec, ho, xag);
  k_mlp   <<<dim3(NP, BB), 32, 0, stream>>>(ho, E, TD, TW, fc1w, fc1b, fc2w,
                                            fc2b, fc3w, fc3b, DE);
  k_dax   <<<dim3(NP / 16, BB), 32, 0, stream>>>(DE, xpad, xdg2);
  k_final <<<NN, 256, 0, stream>>>(xag, xdg2, E, Wp, biasp, Wdg, pa, pb,
                                   (float*)d_out);
}